// CTProjector3D_50955491999807
// MI455X (gfx1250) — compile-verified
//
#include <hip/hip_runtime.h>

typedef float v2f __attribute__((ext_vector_type(2)));
typedef float v8f __attribute__((ext_vector_type(8)));

#define NXD 256
#define NYD 256
#define NZD 256

// ---------------------------------------------------------------------------
// Kernel 0: 3x3 matrix inverse (cofactor method), one thread. Writes 9 floats.
// ---------------------------------------------------------------------------
__global__ void ct_inv3_kernel(const float* __restrict__ M, float* __restrict__ minv) {
    if (threadIdx.x == 0 && blockIdx.x == 0) {
        float a = M[0], b = M[1], c = M[2];
        float d = M[3], e = M[4], f = M[5];
        float g = M[6], h = M[7], i = M[8];
        float A =  (e * i - f * h);
        float B = -(d * i - f * g);
        float C =  (d * h - e * g);
        float det = a * A + b * B + c * C;
        float r = 1.0f / det;
        minv[0] = A * r;  minv[1] = -(b * i - c * h) * r;  minv[2] =  (b * f - c * e) * r;
        minv[3] = B * r;  minv[4] =  (a * i - c * g) * r;  minv[5] = -(a * f - c * d) * r;
        minv[6] = C * r;  minv[7] = -(a * h - b * g) * r;  minv[8] =  (a * e - b * d) * r;
    }
}

// ---------------------------------------------------------------------------
// Kernel 1: per-ray transform as a real GEMM on the matrix core.
//   q = Minv*(src - b), d = Minv*direc via V_WMMA_F32_16X16X4_F32.
// One wave32 per 16 rays. A = Minv padded to 16x4, B = 4x16 point block.
// A layout (16x4 f32): lane m<16 -> row m; VGPR0 = K(2*half), VGPR1 = K(2*half+1)
// B layout (4x16 f32): VGPR0 lanes0-15 = row0, lanes16-31 = row2;
//                      VGPR1 lanes0-15 = row1, lanes16-31 = row3 (zero pad).
// D (16x16 f32): lane n<16, VGPR v -> D[v][n]; rows 0..2 = transformed xyz.
// ---------------------------------------------------------------------------
__global__ void ct_transform_kernel(const float* __restrict__ src,
                                    const float* __restrict__ dst,
                                    const float* __restrict__ bvec,
                                    const float* __restrict__ minv,
                                    float4* __restrict__ q4,
                                    float4* __restrict__ d4,
                                    int n_ray) {
    int wave = (blockIdx.x * blockDim.x + threadIdx.x) >> 5;
    int lane = threadIdx.x & 31;
    int half = lane >> 4;        // 0: lanes 0-15, 1: lanes 16-31
    int m    = lane & 15;
    int base = wave * 16;
    if (base >= n_ray) return;   // wave-uniform: EXEC stays all-ones for WMMA

    int ray = base + m;
    if (ray >= n_ray) ray = n_ray - 1;   // clamp (n_ray is a multiple of 16 anyway)

    // A-matrix: Minv padded with zeros to 16x4
    v2f A;
    A.x = (m < 3) ? minv[m * 3 + 2 * half] : 0.0f;              // K = 0 or 2
    A.y = (m < 3 && half == 0) ? minv[m * 3 + 1] : 0.0f;        // K = 1 (K=3 is pad)

    float bx = bvec[0], by = bvec[1], bz = bvec[2];
    float sx = src[ray * 3 + 0], sy = src[ray * 3 + 1], sz = src[ray * 3 + 2];
    float ex = dst[ray * 3 + 0], ey = dst[ray * 3 + 1], ez = dst[ray * 3 + 2];
    float vx = ex - sx, vy = ey - sy, vz = ez - sz;

    // B-matrices: column m = ray's point, rows = components (row 3 = 0 pad)
    v2f Bq, Bd;
    Bq.x = half ? (sz - bz) : (sx - bx);
    Bq.y = half ? 0.0f      : (sy - by);
    Bd.x = half ? vz : vx;
    Bd.y = half ? 0.0f : vy;

    v8f Z = {0.0f, 0.0f, 0.0f, 0.0f, 0.0f, 0.0f, 0.0f, 0.0f};
    // 8-arg form: (neg_a, A, neg_b, B, c_mod, C, reuse_a, reuse_b)
    v8f Q = __builtin_amdgcn_wmma_f32_16x16x4_f32(false, A, false, Bq, (short)0, Z, false, false);
    v8f D = __builtin_amdgcn_wmma_f32_16x16x4_f32(false, A, false, Bd, (short)0, Z, false, false);

    if (half == 0) {
        float nrm = sqrtf(vx * vx + vy * vy + vz * vz);   // |dst - src|
        q4[ray] = make_float4(Q[0], Q[1], Q[2], nrm);
        d4[ray] = make_float4(D[0], D[1], D[2], 0.0f);
    }
}

// ---------------------------------------------------------------------------
// Kernel 2: main projector. One wave32 per ray; lanes stride the K-1 segments
// so t_sorted loads are coalesced. Volume (64 MB) is L2-resident on MI455X.
// Per segment: 3 FMA + v_rndne round (matches jnp.round half-to-even) +
// bounds check + predicated gather + FMA accumulate. Wave32 butterfly reduce.
// ---------------------------------------------------------------------------
__global__ void ct_project_kernel(const float* __restrict__ volume,
                                  const float* __restrict__ tsort,
                                  const float4* __restrict__ q4,
                                  const float4* __restrict__ d4,
                                  float* __restrict__ out,
                                  int n_ray, int K) {
    int wave = blockIdx.x * (blockDim.x >> 5) + (threadIdx.x >> 5);
    int lane = threadIdx.x & 31;
    if (wave >= n_ray) return;

    const float4 q  = q4[wave];   // (qx,qy,qz, |direc|)
    const float4 dv = d4[wave];   // (dx,dy,dz, --)
    const float* tr = tsort + (size_t)wave * (size_t)K;

    float acc = 0.0f;
    for (int k = lane; k < K - 1; k += 32) {
        float t0 = tr[k];
        float t1 = tr[k + 1];
        float tm = 0.5f * (t0 + t1);
        float fi = fmaf(tm, dv.x, q.x);
        float fj = fmaf(tm, dv.y, q.y);
        float fk = fmaf(tm, dv.z, q.z);
        int ii = __float2int_rn(fi);   // v_cvt round-to-nearest-even
        int jj = __float2int_rn(fj);
        int kk = __float2int_rn(fk);
        bool valid = ((unsigned)ii < (unsigned)NXD) &
                     ((unsigned)jj < (unsigned)NYD) &
                     ((unsigned)kk < (unsigned)NZD);
        if (valid) {
            float vox = volume[((size_t)ii * NYD + (size_t)jj) * NZD + (size_t)kk];
            acc = fmaf(vox, (t1 - t0) * q.w, acc);
        }
    }

    // wave32 reduction (CDNA5 is wave32-only; do NOT hardcode 64)
    #pragma unroll
    for (int off = 16; off > 0; off >>= 1)
        acc += __shfl_xor(acc, off, 32);

    if (lane == 0) out[wave] = acc;
}

// ---------------------------------------------------------------------------
// Host launcher
// Inputs (setup_inputs order): 0 volume, 1 M, 2 b, 3 src, 4 dst, 5 t_sorted
// Workspace layout: [0..15] floats: Minv (9 used); then q4[n_ray]; d4[n_ray].
// ---------------------------------------------------------------------------
extern "C" void kernel_launch(void* const* d_in, const int* in_sizes, int n_in,
                              void* d_out, int out_size, void* d_ws, size_t ws_size,
                              hipStream_t stream) {
    const float* volume = (const float*)d_in[0];
    const float* M      = (const float*)d_in[1];
    const float* b      = (const float*)d_in[2];
    const float* src    = (const float*)d_in[3];
    const float* dst    = (const float*)d_in[4];
    const float* tsort  = (const float*)d_in[5];

    int n_ray = in_sizes[3] / 3;
    int K     = in_sizes[5] / n_ray;

    float*  ws   = (float*)d_ws;
    float*  minv = ws;
    float4* q4   = (float4*)(ws + 16);      // 64B offset, 16B aligned
    float4* d4   = q4 + n_ray;

    ct_inv3_kernel<<<1, 32, 0, stream>>>(M, minv);

    const int wavesPerBlock = 8;            // 256 threads = 8 wave32
    int tiles   = (n_ray + 15) / 16;
    int blocks1 = (tiles + wavesPerBlock - 1) / wavesPerBlock;
    ct_transform_kernel<<<blocks1, 256, 0, stream>>>(src, dst, b, minv, q4, d4, n_ray);

    int blocks2 = (n_ray + wavesPerBlock - 1) / wavesPerBlock;
    ct_project_kernel<<<blocks2, 256, 0, stream>>>(volume, tsort, q4, d4,
                                                   (float*)d_out, n_ray, K);
}